// GCNModel_51049981280880
// MI455X (gfx1250) — compile-verified
//
#include <hip/hip_runtime.h>
#include <hip/hip_bf16.h>
#include <stdint.h>

#define IN_DIM  128
#define HID_DIM 128
#define OUT_DIM 64

typedef __attribute__((ext_vector_type(2))) float v2f;
typedef __attribute__((ext_vector_type(8))) float v8f;

// ---------------------------------------------------------------------------
// Guaranteed-hardware float atomic add (GLOBAL_ATOMIC_ADD_F32, non-returning).
// Avoids any chance of a CAS-loop expansion; completion is tracked by STOREcnt
// and drained by S_ENDPGM's implicit wait-idle.
// ---------------------------------------------------------------------------
__device__ __forceinline__ void gatomic_add_f32(float* p, float v) {
  asm volatile("global_atomic_add_f32 %0, %1, off"
               :
               : "v"((unsigned long long)(uintptr_t)p), "v"(v)
               : "memory");
}

// ---------------------------------------------------------------------------
// WMMA fp32 GEMM with fused GCN epilogue:
//   XW  = act(A) @ B                      (act = ReLU for layers 2/3)
//   OUT = dis^2 * XW + bias               (self-loop term + bias, init base)
// One wave per 16x16 tile via V_WMMA_F32_16X16X4_F32; blockDim=(32,4).
// Wave-uniform guards keep EXEC all-1s for WMMA.
// ---------------------------------------------------------------------------
template <int K, int DOUT, bool RELU>
__global__ __launch_bounds__(128)
void gemm_wmma_f32(const float* __restrict__ A, const float* __restrict__ B,
                   const float* __restrict__ dis, const float* __restrict__ bias,
                   float* __restrict__ XW, float* __restrict__ OUT, int N) {
  const int lane = threadIdx.x & 31;
  const int m0 = blockIdx.x * 16;
  const int n0 = (blockIdx.y * 4 + threadIdx.y) * 16;
  if (m0 >= N || n0 >= DOUT) return;

  const int mrow = m0 + (lane & 15);      // A: lane holds row m, K pair per half
  const int ncol = n0 + (lane & 15);      // B/C/D: lane holds column n
  const int khi  = (lane >> 4) * 2;       // lanes 16-31 hold the upper K pair

  const float* arow = A + (size_t)mrow * K + khi;
  const float* bcol = B + (size_t)khi * DOUT + ncol;

  v8f acc = {};
#pragma unroll 8
  for (int k = 0; k < K; k += 4) {
    v2f a, b;
    a.x = arow[k];
    a.y = arow[k + 1];
    if (RELU) { a.x = fmaxf(a.x, 0.0f); a.y = fmaxf(a.y, 0.0f); }
    b.x = bcol[k * DOUT];
    b.y = bcol[(k + 1) * DOUT];
    acc = __builtin_amdgcn_wmma_f32_16x16x4_f32(false, a, false, b,
                                                (short)0, acc, false, false);
  }

  // D layout: VGPR v holds row m = v + 8*(lane>>4), col = lane&15
  const int mbase = m0 + 8 * (lane >> 4);
  const float bn = bias[ncol];
  float* xwp = XW  + (size_t)mbase * DOUT + ncol;
  float* op  = OUT + (size_t)mbase * DOUT + ncol;
#pragma unroll
  for (int v = 0; v < 8; ++v) {
    float c = acc[v];
    float s = dis[mbase + v];
    xwp[(size_t)v * DOUT] = c;
    op [(size_t)v * DOUT] = s * s * c + bn;
  }
}

// ---------------------------------------------------------------------------
// Degree / normalization
// ---------------------------------------------------------------------------
__global__ void deg_init_kernel(float* __restrict__ deg, int N) {
  int i = blockIdx.x * blockDim.x + threadIdx.x;
  if (i < N) deg[i] = 1.0f;                       // self-loop weight
}

__global__ void deg_accum_kernel(const int* __restrict__ dst,
                                 const float* __restrict__ ew,
                                 float* __restrict__ deg, int E) {
  int e = blockIdx.x * blockDim.x + threadIdx.x;
  if (e < E) gatomic_add_f32(&deg[dst[e]], ew[e]);
}

__global__ void rsqrt_kernel(float* __restrict__ deg, int N) {
  int i = blockIdx.x * blockDim.x + threadIdx.x;
  if (i < N) deg[i] = rsqrtf(deg[i]);             // deg >= 1, safe
}

// ---------------------------------------------------------------------------
// Edge aggregation: out[dst] += dis[src]*ew*dis[dst] * xw[src]
// One 32-lane group per edge; each lane handles D/32 consecutive floats
// (float4 slices for D=128, float2 for D=64). Atomics resolve in L2
// (xw + out working set ~77 MB << 192 MB L2).
// ---------------------------------------------------------------------------
template <int D>
__global__ __launch_bounds__(256)
void edge_agg_kernel(const int* __restrict__ src, const int* __restrict__ dst,
                     const float* __restrict__ ew, const float* __restrict__ dis,
                     const float* __restrict__ xw, float* __restrict__ out, int E) {
  constexpr int V = D / 32;                       // floats per lane
  const int lane = threadIdx.x & 31;
  const int e = blockIdx.x * 8 + (threadIdx.x >> 5);
  if (e >= E) return;

  const int s = src[e];
  const int t = dst[e];
  const float norm = dis[s] * ew[e] * dis[t];

  const float* xp = xw + (size_t)s * D + lane * V;
  float* op = out + (size_t)t * D + lane * V;

  float vals[V];
  if constexpr (V == 4) {
    float4 v4 = *(const float4*)xp;
    vals[0] = v4.x; vals[1] = v4.y; vals[2] = v4.z; vals[3] = v4.w;
  } else {
    float2 v2 = *(const float2*)xp;
    vals[0] = v2.x; vals[1] = v2.y;
  }
#pragma unroll
  for (int j = 0; j < V; ++j) gatomic_add_f32(&op[j], norm * vals[j]);
}

// ---------------------------------------------------------------------------
extern "C" void kernel_launch(void* const* d_in, const int* in_sizes, int n_in,
                              void* d_out, int out_size, void* d_ws, size_t ws_size,
                              hipStream_t stream) {
  const float* x  = (const float*)d_in[0];
  const int*   ei = (const int*)  d_in[1];
  const float* ew = (const float*)d_in[2];
  const float* W1 = (const float*)d_in[3];
  const float* b1 = (const float*)d_in[4];
  const float* W2 = (const float*)d_in[5];
  const float* b2 = (const float*)d_in[6];
  const float* W3 = (const float*)d_in[7];
  const float* b3 = (const float*)d_in[8];
  float* out = (float*)d_out;

  const int N = in_sizes[0] / IN_DIM;
  const int E = in_sizes[1] / 2;
  const int* src = ei;
  const int* dst = ei + E;

  // Workspace layout: h[N*128] | xw[N*128] | dis[N]
  float* h   = (float*)d_ws;
  float* xw  = h  + (size_t)N * HID_DIM;
  float* dis = xw + (size_t)N * HID_DIM;

  const int T = 256;

  // --- symmetric normalization: dis = rsqrt(sum_dst(ew) + 1) ---
  deg_init_kernel<<<(N + T - 1) / T, T, 0, stream>>>(dis, N);
  deg_accum_kernel<<<(E + T - 1) / T, T, 0, stream>>>(dst, ew, dis, E);
  rsqrt_kernel<<<(N + T - 1) / T, T, 0, stream>>>(dis, N);

  const dim3 gblock(32, 4);
  const int mtiles = (N + 15) / 16;

  // --- layer 1: xw = x@W1; h = dis^2*xw + b1; h += edge agg; (ReLU fused fwd) ---
  gemm_wmma_f32<IN_DIM, HID_DIM, false>
      <<<dim3(mtiles, HID_DIM / 64), gblock, 0, stream>>>(x, W1, dis, b1, xw, h, N);
  edge_agg_kernel<HID_DIM><<<(E + 7) / 8, T, 0, stream>>>(src, dst, ew, dis, xw, h, E);

  // --- layer 2: xw = relu(h)@W2; h = dis^2*xw + b2; h += edge agg ---
  gemm_wmma_f32<HID_DIM, HID_DIM, true>
      <<<dim3(mtiles, HID_DIM / 64), gblock, 0, stream>>>(h, W2, dis, b2, xw, h, N);
  edge_agg_kernel<HID_DIM><<<(E + 7) / 8, T, 0, stream>>>(src, dst, ew, dis, xw, h, E);

  // --- layer 3: xw = relu(h)@W3; out = dis^2*xw + b3; out += edge agg ---
  gemm_wmma_f32<HID_DIM, OUT_DIM, true>
      <<<dim3(mtiles, 1), gblock, 0, stream>>>(h, W3, dis, b3, xw, out, N);
  edge_agg_kernel<OUT_DIM><<<(E + 7) / 8, T, 0, stream>>>(src, dst, ew, dis, xw, out, E);
}